// GraphCounteRGAN_86990267613262
// MI455X (gfx1250) — compile-verified
//
#include <hip/hip_runtime.h>
#include <stdint.h>

// ---------------- problem constants (match reference) ----------------
#define N_NODES  50000
#define N_FEAT   128
#define HID      64
#define HID2     32
#define N_EDGES  1600000

typedef float v2f __attribute__((ext_vector_type(2)));
typedef float v8f __attribute__((ext_vector_type(8)));

// ---------------- workspace layout (in floats, all 16-float aligned) ----------------
static constexpr long long OFF_DINV = 0;                                      // 50000 (deg then dinv, in place)
static constexpr long long OFF_H1   = 50048;                                  // N*64
static constexpr long long OFF_AGG1 = OFF_H1   + (long long)N_NODES * HID;    // N*64
static constexpr long long OFF_H2   = OFF_AGG1 + (long long)N_NODES * HID;    // N*32
static constexpr long long OFF_AGG2 = OFF_H2   + (long long)N_NODES * HID2;   // N*32
static constexpr long long OFF_XR   = OFF_AGG2 + (long long)N_NODES * HID2;   // N*128
static constexpr long long OFF_HD   = OFF_XR   + (long long)N_NODES * N_FEAT; // N*2
static constexpr long long OFF_AGGD = OFF_HD   + (long long)N_NODES * 2;      // N*2
static constexpr long long OFF_ACC  = OFF_AGGD + (long long)N_NODES * 2;      // 16 (padded)
static constexpr long long OFF_EREC = OFF_ACC  + 16;                          // E*4 (packed edge records)

// ---------------- degree / normalization ----------------
__global__ void deg_init(float* __restrict__ deg) {
    int i = blockIdx.x * blockDim.x + threadIdx.x;
    if (i < N_NODES) deg[i] = 1.0f;  // self-loop weight
}

__global__ void deg_accum(const long long* __restrict__ dst,
                          const float* __restrict__ ew,
                          float* __restrict__ deg) {
    int e = blockIdx.x * blockDim.x + threadIdx.x;
    if (e < N_EDGES) atomicAdd(&deg[(int)dst[e]], ew[e]);
}

__global__ void deg_to_dinv(float* __restrict__ deg) {
    int i = blockIdx.x * blockDim.x + threadIdx.x;
    if (i < N_NODES) {
        float d = deg[i];
        deg[i] = (d > 0.0f) ? rsqrtf(d) : 0.0f;
    }
}

// ---------------- packed edge records: {src, dst, norm, pad} = 16B ----------------
// norm = dinv[src]*w*dinv[dst] is layer-invariant -> compute once, reuse 3x.
// Scatter kernels then issue only 2 loads/thread (record b128 + h-row b128).
__global__ void edge_prep(const long long* __restrict__ src,
                          const long long* __restrict__ dst,
                          const float* __restrict__ ew,
                          const float* __restrict__ dinv,
                          float4* __restrict__ rec) {
    int e = blockIdx.x * blockDim.x + threadIdx.x;
    if (e >= N_EDGES) return;
    const int s = (int)src[e];
    const int d = (int)dst[e];
    const float norm = dinv[s] * ew[e] * dinv[d];
    rec[e] = make_float4(__int_as_float(s), __int_as_float(d), norm, 0.0f);
}

// ---------------- fp32 WMMA GEMM: D[M,C] = A[M,K] @ B[K,C] ----------------
// Compile-time K, C. One wave computes a full 16-row x C strip (all column
// tiles): the A fragment is loaded once (global_load_b64, immediate offsets)
// and reused across CT back-to-back V_WMMA_F32_16X16X4_F32 ops. B is
// cooperatively staged into LDS, zero-padded to CT*16 columns, stored as
// (k,k+1) float2 pairs so each lane's B fragment is one ds_load_b64
// (compiler merges pairs into ds_load_2addr_b64).
// Fragment layouts (ISA 7.12.2):
//   A(16x4):  lane l holds M=l%16, K = k + 2*(l>=16) + {0,1}
//   B(4x16):  lane l holds N=l%16, K = k + 2*(l>=16) + {0,1}
//   D(16x16): reg r holds row r + 8*(l>=16), col l%16
template <int K, int C, bool RELU_A>
__global__ void wmma_gemm_f32(const float* __restrict__ A,
                              const float* __restrict__ B,
                              float* __restrict__ D) {
    constexpr int CT = (C + 15) / 16;  // column tiles
    constexpr int CW = CT * 16;        // padded column width
    __shared__ float2 ldsB[(K / 2) * CW];

    // cooperative, zero-padded stage of B into LDS
    for (int idx = threadIdx.x; idx < (K / 2) * CW; idx += blockDim.x) {
        const int p   = idx / CW;   // k-pair index
        const int col = idx % CW;
        const int k   = p * 2;
        float2 v;
        v.x = (col < C) ? B[k * C + col]       : 0.0f;
        v.y = (col < C) ? B[(k + 1) * C + col] : 0.0f;
        ldsB[idx] = v;
    }
    __syncthreads();

    const int lane = threadIdx.x & 31;
    const int wave = threadIdx.x >> 5;
    const int rowTile = blockIdx.x * (blockDim.x >> 5) + wave;
    if (rowTile >= N_NODES / 16) return;  // wave-uniform; EXEC all-ones for WMMA

    const int m     = lane & 15;            // A row within tile
    const int n     = lane & 15;            // B/D col within tile
    const int khalf = (lane >> 4) << 1;     // 0 for lanes 0-15, 2 for lanes 16-31

    const float* __restrict__ arow =
        A + (long long)(rowTile * 16 + m) * K + khalf;

    v8f acc[CT];
#pragma unroll
    for (int ct = 0; ct < CT; ++ct) acc[ct] = {};

#pragma unroll
    for (int k = 0; k < K; k += 4) {
        float2 av = *reinterpret_cast<const float2*>(arow + k);  // b64, imm offset
        v2f a;
        a[0] = RELU_A ? fmaxf(av.x, 0.0f) : av.x;
        a[1] = RELU_A ? fmaxf(av.y, 0.0f) : av.y;
        const int prow = ((k + khalf) >> 1) * CW;
#pragma unroll
        for (int ct = 0; ct < CT; ++ct) {
            float2 bv = ldsB[prow + ct * 16 + n];                // ds_load_b64
            v2f b;
            b[0] = bv.x;
            b[1] = bv.y;
            acc[ct] = __builtin_amdgcn_wmma_f32_16x16x4_f32(
                false, a, false, b, (short)0, acc[ct], false, false);
        }
    }

    const int mbase = rowTile * 16 + ((lane >> 4) << 3);  // +8 rows, upper half-wave
#pragma unroll
    for (int ct = 0; ct < CT; ++ct) {
        const int col = ct * 16 + n;
        if (col < C) {  // only matters for the padded C=2 discriminator GEMM
#pragma unroll
            for (int r = 0; r < 8; ++r)
                D[(long long)(mbase + r) * C + col] = acc[ct][r];
        }
    }
}

// ---------------- aggregation: self-loop + bias init ----------------
template <int C>
__global__ void agg_init(const float* __restrict__ h,
                         const float* __restrict__ dinv,
                         const float* __restrict__ bias,
                         float* __restrict__ agg) {
    long long i = (long long)blockIdx.x * blockDim.x + threadIdx.x;
    if (i >= (long long)N_NODES * C) return;
    const int node = (int)(i / C);
    const int c    = (int)(i % C);
    const float di = dinv[node];
    agg[i] = bias[c] + h[i] * di * di;  // self-loop: norm = dinv*1*dinv
}

// ---------------- aggregation: edge scatter (L2-resident f32 atomics) ----------------
// idx = e * (C/VEC) + q : consecutive threads cover consecutive channel chunks
// of the same edge -> fully coalesced b128 gathers of h[src] rows. Atomics
// without return are fire-and-forget (STOREcnt), so the kernel is issue
// bound: only 2 loads per thread (packed edge record + h chunk).
template <int C, int VEC>
__global__ void agg_edges(const float* __restrict__ h,
                          const float4* __restrict__ rec,
                          float* __restrict__ agg) {
    constexpr int TPE = C / VEC;
    long long idx = (long long)blockIdx.x * blockDim.x + threadIdx.x;
    if (idx >= (long long)N_EDGES * TPE) return;
    const int e = (int)(idx / TPE);   // TPE is a power of two -> shifts
    const int q = (int)(idx % TPE);

    const float4 r = rec[e];          // global_load_b128 (same line across TPE lanes)
    const int s = __float_as_int(r.x);
    const int d = __float_as_int(r.y);
    const float norm = r.z;

    const float* __restrict__ hp = h   + (long long)s * C + q * VEC;
    float* __restrict__       ap = agg + (long long)d * C + q * VEC;

    if constexpr (VEC == 4) {
        float4 hv = *reinterpret_cast<const float4*>(hp);  // global_load_b128
        atomicAdd(ap + 0, hv.x * norm);
        atomicAdd(ap + 1, hv.y * norm);
        atomicAdd(ap + 2, hv.z * norm);
        atomicAdd(ap + 3, hv.w * norm);
    } else {
        float2 hv = *reinterpret_cast<const float2*>(hp);
        atomicAdd(ap + 0, hv.x * norm);
        atomicAdd(ap + 1, hv.y * norm);
    }
}

// ---------------- residual: x = tile(tanh(agg2), 1x4) + features ----------------
__global__ void residual_tile(const float* __restrict__ agg2,
                              const float* __restrict__ feat,
                              float* __restrict__ xr) {
    long long i = (long long)blockIdx.x * blockDim.x + threadIdx.x;
    if (i >= (long long)N_NODES * N_FEAT) return;
    const int node = (int)(i / N_FEAT);
    const int c    = (int)(i % N_FEAT);
    xr[i] = tanhf(agg2[(long long)node * HID2 + (c & (HID2 - 1))]) + feat[i];
}

// ---------------- final reduction: sigmoid(sum relu(d)*w_fc + b_fc) ----------------
__global__ void acc_zero(float* __restrict__ acc) {
    if (threadIdx.x == 0 && blockIdx.x == 0) acc[0] = 0.0f;
}

__global__ void disc_reduce(const float* __restrict__ aggd,
                            const float* __restrict__ w_fc,
                            float* __restrict__ acc) {
    __shared__ float sdata[256];
    const int tid = threadIdx.x;
    long long i = (long long)blockIdx.x * blockDim.x + tid;
    float v = 0.0f;
    if (i < (long long)N_NODES * 2)
        v = fmaxf(aggd[i], 0.0f) * w_fc[i];
    sdata[tid] = v;
    __syncthreads();
#pragma unroll
    for (int s = 128; s > 0; s >>= 1) {
        if (tid < s) sdata[tid] += sdata[tid + s];
        __syncthreads();
    }
    if (tid == 0) atomicAdd(acc, sdata[0]);
}

__global__ void final_sigmoid(const float* __restrict__ acc,
                              const float* __restrict__ b_fc,
                              float* __restrict__ out) {
    if (threadIdx.x == 0 && blockIdx.x == 0) {
        const float logit = acc[0] + b_fc[0];
        out[0] = 1.0f / (1.0f + expf(-logit));
    }
}

// ---------------- launch ----------------
static inline int cdiv(long long a, long long b) { return (int)((a + b - 1) / b); }

extern "C" void kernel_launch(void* const* d_in, const int* in_sizes, int n_in,
                              void* d_out, int out_size, void* d_ws, size_t ws_size,
                              hipStream_t stream) {
    (void)in_sizes; (void)n_in; (void)out_size; (void)ws_size;

    const float*     feat = (const float*)d_in[0];
    const long long* ei   = (const long long*)d_in[1];   // int64 [2, E]
    const float*     ew   = (const float*)d_in[2];
    const float*     W1   = (const float*)d_in[3];
    const float*     b1   = (const float*)d_in[4];
    const float*     W2   = (const float*)d_in[5];
    const float*     b2   = (const float*)d_in[6];
    const float*     Wd   = (const float*)d_in[7];
    const float*     bd   = (const float*)d_in[8];
    const float*     wfc  = (const float*)d_in[9];
    const float*     bfc  = (const float*)d_in[10];
    float*           out  = (float*)d_out;

    const long long* src = ei;
    const long long* dst = ei + N_EDGES;

    float*  ws   = (float*)d_ws;
    float*  dinv = ws + OFF_DINV;
    float*  h1   = ws + OFF_H1;
    float*  agg1 = ws + OFF_AGG1;
    float*  h2   = ws + OFF_H2;
    float*  agg2 = ws + OFF_AGG2;
    float*  xr   = ws + OFF_XR;
    float*  hd   = ws + OFF_HD;
    float*  aggd = ws + OFF_AGGD;
    float*  acc  = ws + OFF_ACC;
    float4* erec = (float4*)(ws + OFF_EREC);

    const int B = 256;                 // 8 wave32 per block
    const int wavesPerBlock = B / 32;
    const int rowTiles = N_NODES / 16; // 3125 (exact)
    const int gemmBlocks = cdiv(rowTiles, wavesPerBlock);

    // --- degree normalization + packed edge records ---
    deg_init<<<cdiv(N_NODES, B), B, 0, stream>>>(dinv);
    deg_accum<<<cdiv(N_EDGES, B), B, 0, stream>>>(dst, ew, dinv);
    deg_to_dinv<<<cdiv(N_NODES, B), B, 0, stream>>>(dinv);
    edge_prep<<<cdiv(N_EDGES, B), B, 0, stream>>>(src, dst, ew, dinv, erec);

    // --- layer 1: h1 = feat @ W1 ; agg1 = GCN-aggregate + b1 ---
    wmma_gemm_f32<N_FEAT, HID, false><<<gemmBlocks, B, 0, stream>>>(feat, W1, h1);
    agg_init<HID><<<cdiv((long long)N_NODES * HID, B), B, 0, stream>>>(h1, dinv, b1, agg1);
    agg_edges<HID, 4><<<cdiv((long long)N_EDGES * (HID / 4), B), B, 0, stream>>>(
        h1, erec, agg1);

    // --- layer 2: h2 = relu(agg1) @ W2 ; agg2 = GCN-aggregate + b2 ---
    wmma_gemm_f32<HID, HID2, true><<<gemmBlocks, B, 0, stream>>>(agg1, W2, h2);
    agg_init<HID2><<<cdiv((long long)N_NODES * HID2, B), B, 0, stream>>>(h2, dinv, b2, agg2);
    agg_edges<HID2, 4><<<cdiv((long long)N_EDGES * (HID2 / 4), B), B, 0, stream>>>(
        h2, erec, agg2);

    // --- residual: xr = tile(tanh(agg2), 4) + feat ---
    residual_tile<<<cdiv((long long)N_NODES * N_FEAT, B), B, 0, stream>>>(agg2, feat, xr);

    // --- discriminator: hd = xr @ Wd (C=2, column-padded WMMA) ; aggregate + bd ---
    wmma_gemm_f32<N_FEAT, 2, false><<<gemmBlocks, B, 0, stream>>>(xr, Wd, hd);
    agg_init<2><<<cdiv((long long)N_NODES * 2, B), B, 0, stream>>>(hd, dinv, bd, aggd);
    agg_edges<2, 2><<<cdiv((long long)N_EDGES * 1, B), B, 0, stream>>>(hd, erec, aggd);

    // --- relu -> dot(w_fc) -> + b_fc -> sigmoid ---
    acc_zero<<<1, 1, 0, stream>>>(acc);
    disc_reduce<<<cdiv((long long)N_NODES * 2, B), B, 0, stream>>>(aggd, wfc, acc);
    final_sigmoid<<<1, 1, 0, stream>>>(acc, bfc, out);
}